// CustomEmbeddings_51084341019231
// MI455X (gfx1250) — compile-verified
//
#include <hip/hip_runtime.h>
#include <hip/hip_bf16.h>
#include <stddef.h>

// ---------------- problem constants ----------------
#define VOCAB      32100
#define DIM        3200          // d
#define HID        6400          // 2d
#define NTOK       16384         // 8 * 2048
#define NEW_START  32000

#define PASS_TOK   256           // tokens handled per (gemm1, gemm2) pass
#define NPASS      (NTOK / PASS_TOK)   // 64 worst-case passes

// ---------------- WMMA vector types ----------------
typedef __attribute__((ext_vector_type(2))) float v2f;
typedef __attribute__((ext_vector_type(8))) float v8f;

// ---------------- workspace layout -----------------
// [0, 4)          : int counter (number of masked tokens)
// [64, ...)       : int indices[NTOK] (compacted masked positions)
// [WS_OFF_ZROW)   : float zrow[DIM]  -- all-zero A row for padded lanes
// [WS_OFF_HBUF)   : float hbuf[PASS_TOK * HID]  (relu(x@w1+b1))
#define WS_OFF_IDX   64
#define WS_OFF_ZROW  (WS_OFF_IDX + NTOK * 4)            // 65600
#define WS_OFF_HBUF  (WS_OFF_ZROW + DIM * 4)            // 78400 (64-aligned)

__global__ void k_init(int* counter, float* zrow) {
    int i = blockIdx.x * blockDim.x + threadIdx.x;
    if (i == 0) *counter = 0;
    if (i < DIM) zrow[i] = 0.0f;
}

// compact positions with id >= NEW_START
__global__ void k_build_index(const int* __restrict__ ids, int* counter,
                              int* __restrict__ indices) {
    int i = blockIdx.x * blockDim.x + threadIdx.x;
    if (i < NTOK && ids[i] >= NEW_START) {
        int p = atomicAdd(counter, 1);
        indices[p] = i;
    }
}

// out[row] = emb_table[ids[row]]  -- pure bandwidth, float4 copies
__global__ void k_gather_rows(const int* __restrict__ ids,
                              const float* __restrict__ emb,
                              float* __restrict__ out) {
    int row = blockIdx.x;
    int id  = ids[row];
    const float4* __restrict__ src = (const float4*)(emb + (size_t)id * DIM);
    float4* __restrict__ dst       = (float4*)(out + (size_t)row * DIM);
    for (int j = threadIdx.x; j < DIM / 4; j += blockDim.x)
        dst[j] = src[j];
}

// GEMM1: hbuf[t_local, :] = relu(emb[ids[indices[t]]] @ w1 + b1)
// grid: (16 row-tiles, HID/16/8 wave-groups), block 256 (8 wave32 -> 8 col tiles)
__global__ void k_gemm1(const int* __restrict__ ids,
                        const int* __restrict__ indices,
                        const int* __restrict__ counter,
                        const float* __restrict__ emb,
                        const float* __restrict__ w1,
                        const float* __restrict__ b1,
                        const float* __restrict__ zrow,
                        float* __restrict__ hbuf,
                        int tokenBase) {
    const int count = *counter;
    const int rowTile = blockIdx.x;                 // 0..15 within pass
    const int t0 = tokenBase + rowTile * 16;
    if (t0 >= count) return;                        // whole tile beyond mask count

    const int wave = threadIdx.x >> 5;
    const int lane = threadIdx.x & 31;
    const int colTile = blockIdx.y * 8 + wave;      // 0..(HID/16 - 1)
    const int n0 = colTile * 16;
    const int lm = lane & 15;                       // N (B/C/D) or M (A)
    const int ko = (lane >> 4) * 2;                 // A/B K sub-offset per lane half

    // A-matrix row for this lane: real embedding row, or the zero row for
    // padded lanes.  Selecting the POINTER once keeps the K-loop loads
    // unconditional (no divergent exec-mask juggling in the hot loop).
    const int t = t0 + lm;
    const float* __restrict__ aRow =
        (t < count) ? (emb + (size_t)ids[indices[t]] * DIM) : zrow;

    const float* __restrict__ bCol = w1 + n0 + lm;

    v8f c = {};
    #pragma unroll 4
    for (int k = 0; k < DIM; k += 4) {
        const int ka = k + ko;
        v2f a = *(const v2f*)(aRow + ka);           // one global_load_b64
        v2f b;
        b.x = bCol[(size_t)ka * HID];
        b.y = bCol[(size_t)(ka + 1) * HID];
        c = __builtin_amdgcn_wmma_f32_16x16x4_f32(
                false, a, false, b, (short)0, c, false, false);
    }

    const float bias = b1[n0 + lm];
    float* __restrict__ hCol = hbuf + (size_t)(rowTile * 16) * HID + n0 + lm;
    #pragma unroll
    for (int j = 0; j < 8; ++j) {
        const int m = j + (lane >> 4) * 8;          // C/D row within tile
        float v = c[j] + bias;
        v = v > 0.0f ? v : 0.0f;                    // ReLU
        hCol[(size_t)m * HID] = v;
    }
}

// GEMM2: out[indices[t], :] = hbuf[t_local, :] @ w2 + b2  (scatter-store)
// grid: (16 row-tiles, DIM/16/8 wave-groups), block 256
__global__ void k_gemm2(const int* __restrict__ indices,
                        const int* __restrict__ counter,
                        const float* __restrict__ hbuf,
                        const float* __restrict__ w2,
                        const float* __restrict__ b2,
                        float* __restrict__ out,
                        int tokenBase) {
    const int count = *counter;
    const int rowTile = blockIdx.x;
    const int t0 = tokenBase + rowTile * 16;
    if (t0 >= count) return;

    const int wave = threadIdx.x >> 5;
    const int lane = threadIdx.x & 31;
    const int colTile = blockIdx.y * 8 + wave;      // 0..(DIM/16 - 1)
    const int n0 = colTile * 16;
    const int lm = lane & 15;
    const int ko = (lane >> 4) * 2;

    const float* __restrict__ aRow =
        hbuf + (size_t)(rowTile * 16 + lm) * HID;   // A row = local h row
    const float* __restrict__ bCol = w2 + n0 + lm;

    v8f c = {};
    #pragma unroll 4
    for (int k = 0; k < HID; k += 4) {
        const int ka = k + ko;
        v2f a = *(const v2f*)(aRow + ka);           // one global_load_b64
        v2f b;
        b.x = bCol[(size_t)ka * DIM];
        b.y = bCol[(size_t)(ka + 1) * DIM];
        c = __builtin_amdgcn_wmma_f32_16x16x4_f32(
                false, a, false, b, (short)0, c, false, false);
    }

    const float bias = b2[n0 + lm];
    #pragma unroll
    for (int j = 0; j < 8; ++j) {
        const int m = j + (lane >> 4) * 8;
        const int t = t0 + m;
        if (t < count) {
            const int pos = indices[t];             // original token position
            out[(size_t)pos * DIM + n0 + lm] = c[j] + bias;
        }
    }
}

extern "C" void kernel_launch(void* const* d_in, const int* in_sizes, int n_in,
                              void* d_out, int out_size, void* d_ws, size_t ws_size,
                              hipStream_t stream) {
    (void)in_sizes; (void)n_in; (void)out_size; (void)ws_size;

    const int*   ids = (const int*)  d_in[0];   // input_ids  [8,2048]
    const float* emb = (const float*)d_in[1];   // emb_table  [32100,3200]
    const float* w1  = (const float*)d_in[2];   // [3200,6400]
    const float* b1  = (const float*)d_in[3];   // [6400]
    const float* w2  = (const float*)d_in[4];   // [6400,3200]
    const float* b2  = (const float*)d_in[5];   // [3200]
    float* out = (float*)d_out;                 // [16384,3200]

    char* ws = (char*)d_ws;
    int*   counter = (int*)ws;
    int*   indices = (int*)(ws + WS_OFF_IDX);
    float* zrow    = (float*)(ws + WS_OFF_ZROW);
    float* hbuf    = (float*)(ws + WS_OFF_HBUF);

    // 1) reset counter + zero row, compact masked token positions
    k_init<<<(DIM + 255) / 256, 256, 0, stream>>>(counter, zrow);
    k_build_index<<<(NTOK + 255) / 256, 256, 0, stream>>>(ids, counter, indices);

    // 2) embedding gather for all positions (bandwidth-dominant path)
    k_gather_rows<<<NTOK, 256, 0, stream>>>(ids, emb, out);

    // 3) masked MLP: fixed 64 passes of 256 tokens; blocks early-exit past count
    dim3 g1(PASS_TOK / 16, HID / 16 / 8);   // (16, 50)
    dim3 g2(PASS_TOK / 16, DIM / 16 / 8);   // (16, 25)
    for (int p = 0; p < NPASS; ++p) {
        const int tokenBase = p * PASS_TOK;
        k_gemm1<<<g1, 256, 0, stream>>>(ids, indices, counter, emb, w1, b1,
                                        zrow, hbuf, tokenBase);
        k_gemm2<<<g2, 256, 0, stream>>>(indices, counter, hbuf, w2, b2,
                                        out, tokenBase);
    }
}